// ConditionalThetaDiagonalSplineLinearXFlowMLP_28544352649526
// MI455X (gfx1250) — compile-verified
//
#include <hip/hip_runtime.h>

typedef __attribute__((ext_vector_type(2))) float v2f;
typedef __attribute__((ext_vector_type(8))) float v8f;

#define NBASIS 5
#define DEGREE 3
#define ROWTILES 8   // 16-row tiles per wave; weight regs amortized across them

// Cubic B-spline basis, open-uniform clamped knots for n=5,p=3:
// knots = [0,0,0,0,0.5,1,1,1,1]. Mirrors reference Cox-de Boor; fully
// unrolled so knot arithmetic constant-folds.
__device__ __forceinline__ void bspline_phi5(float th, float N[8]) {
  float u = fminf(fmaxf(th, 0.0f), 1.0f);
  u = fminf(fmaxf(u, 1e-6f), 1.0f - 1e-6f);
  const float kn[9] = {0.f, 0.f, 0.f, 0.f, 0.5f, 1.f, 1.f, 1.f, 1.f};
#pragma unroll
  for (int i = 0; i < 8; ++i) {
    float span = kn[i + 1] - kn[i];
    N[i] = (u >= kn[i] && u < kn[i + 1] && fabsf(span) >= 1e-15f) ? 1.0f : 0.0f;
  }
#pragma unroll
  for (int r = 1; r <= DEGREE; ++r) {
#pragma unroll
    for (int i = 0; i < 8; ++i) {
      if (i < 8 - r) {
        float d1 = kn[i + r] - kn[i];
        float d2 = kn[i + r + 1] - kn[i + 1];
        float t1 = (fabsf(d1) > 1e-15f) ? (u - kn[i]) / d1 * N[i] : 0.0f;
        float t2 = (fabsf(d2) > 1e-15f) ? (kn[i + r + 1] - u) / d2 * N[i + 1] : 0.0f;
        N[i] = t1 + t2;
      }
    }
  }
}

// One wave owns a 16-column tile and ROWTILES row tiles of 16 rows.
// Coefficient tiles a,b via V_WMMA_F32_16X16X4_F32 (K=5 padded to 8).
// DC>0: compile-time D so all x/out addresses become immediate offsets.
template <int DC>
__global__ __launch_bounds__(256) void spline_affine_wmma_kernel(
    const float* __restrict__ x, const float* __restrict__ theta,
    const float* __restrict__ Wa, const float* __restrict__ ca,
    const float* __restrict__ Wb, const float* __restrict__ cb,
    float* __restrict__ out, int B, int Drt) {
  const int D = (DC > 0) ? DC : Drt;
  const int lane = threadIdx.x & 31;   // wave32
  const int wid  = threadIdx.x >> 5;   // 8 waves per block
  const int half = lane >> 4;
  const int l15  = lane & 15;

  const int colbase = (blockIdx.x * 8 + wid) * 16;   // uniform per wave
  if (colbase >= D) return;
  const int col = colbase + l15;

  // --- Per-wave setup: B-matrices (Wa/Wb columns) + biases.
  // B-matrix 4x16 layout: lanes 0-15 hold K={0,1}, lanes 16-31 hold K={2,3}.
  const int k0 = half ? 2 : 0;
  const int k1 = half ? 3 : 1;
  const float* wa = Wa + (size_t)col * NBASIS;
  const float* wb = Wb + (size_t)col * NBASIS;
  v2f Ba1, Ba2, Bb1, Bb2;
  Ba1.x = wa[k0]; Ba1.y = wa[k1];
  Ba2.x = half ? 0.0f : wa[4]; Ba2.y = 0.0f;   // K=4 live, K=5..7 zero pad
  Bb1.x = wb[k0]; Bb1.y = wb[k1];
  Bb2.x = half ? 0.0f : wb[4]; Bb2.y = 0.0f;
  const float cav = ca[col];
  const float cbv = cb[col];
  const v8f Ca = {cav, cav, cav, cav, cav, cav, cav, cav};
  const v8f Cb = {cbv, cbv, cbv, cbv, cbv, cbv, cbv, cbv};

  const int row0 = blockIdx.y * (ROWTILES * 16);

  // Preload theta for every row tile this wave handles (one wait total).
  float th8[ROWTILES];
#pragma unroll
  for (int t = 0; t < ROWTILES; ++t) {
    const int r = row0 + t * 16 + l15;
    th8[t] = (r < B) ? theta[r] : 0.0f;
  }

  // Single base address; with DC compile-time, every access below is an
  // immediate-offset global_load/store from this base (max offset
  // 119*D*4 = 1.9MB < 2^23 for D=4096).
  const size_t base = (size_t)(row0 + half * 8) * (size_t)D + (size_t)col;
  const float* __restrict__ px   = x + base;
  float* __restrict__       pout = out + base;

#pragma unroll
  for (int t = 0; t < ROWTILES; ++t) {
    const int rowbase = row0 + t * 16;   // uniform per wave
    if (rowbase < B) {
      // Issue all 8 streaming x loads first (distinct regs, no per-elem wait).
      float xv[8];
#pragma unroll
      for (int v = 0; v < 8; ++v)
        xv[v] = __builtin_nontemporal_load(px + (size_t)(t * 16 + v) * D);

      // phi + WMMA coefficient tiles overlap the loads in flight.
      float N[8];
      bspline_phi5(th8[t], N);
      v2f A1, A2;
      A1.x = half ? N[2] : N[0];   // A 16x4 layout: lanes 0-15 K={0,1},
      A1.y = half ? N[3] : N[1];   // lanes 16-31 K={2,3}
      A2.x = half ? 0.0f : N[4];
      A2.y = 0.0f;

      v8f Da = __builtin_amdgcn_wmma_f32_16x16x4_f32(
          false, A1, false, Ba1, (short)0, Ca, false, false);
      Da = __builtin_amdgcn_wmma_f32_16x16x4_f32(
          false, A2, false, Ba2, (short)0, Da, false, false);
      v8f Db = __builtin_amdgcn_wmma_f32_16x16x4_f32(
          false, A1, false, Bb1, (short)0, Cb, false, false);
      Db = __builtin_amdgcn_wmma_f32_16x16x4_f32(
          false, A2, false, Bb2, (short)0, Db, false, false);

#pragma unroll
      for (int v = 0; v < 8; ++v)
        __builtin_nontemporal_store(fmaf(Da[v], xv[v], Db[v]),
                                    pout + (size_t)(t * 16 + v) * D);
    }
  }
}

extern "C" void kernel_launch(void* const* d_in, const int* in_sizes, int n_in,
                              void* d_out, int out_size, void* d_ws, size_t ws_size,
                              hipStream_t stream) {
  const float* x     = (const float*)d_in[0];
  const float* theta = (const float*)d_in[1];
  const float* Wa    = (const float*)d_in[2];
  const float* ca    = (const float*)d_in[3];
  const float* Wb    = (const float*)d_in[4];
  const float* cb    = (const float*)d_in[5];
  float* out = (float*)d_out;

  const int D = in_sizes[3];            // ca has D elements
  const int B = in_sizes[0] / D;        // x is B x D

  const int colTiles = (D + 15) / 16;                 // 256 for D=4096
  const int gx = (colTiles + 7) / 8;                  // 8 column tiles per block
  const int rowTiles = (B + 15) / 16;                 // 1024 for B=16384
  const int gy = (rowTiles + ROWTILES - 1) / ROWTILES;

  dim3 grid(gx, gy, 1);
  dim3 block(256, 1, 1);
  if (D == 4096) {
    hipLaunchKernelGGL((spline_affine_wmma_kernel<4096>), grid, block, 0, stream,
                       x, theta, Wa, ca, Wb, cb, out, B, D);
  } else {
    hipLaunchKernelGGL((spline_affine_wmma_kernel<0>), grid, block, 0, stream,
                       x, theta, Wa, ca, Wb, cb, out, B, D);
  }
}